// parallelAnalyser_26637387170100
// MI455X (gfx1250) — compile-verified
//
#include <hip/hip_runtime.h>
#include <math.h>

// ---------------- problem constants ----------------
#define GXd 24
#define GYd 24
#define GZd 6
#define NCH 10
#define NBATCH 16
#define NCELL 3456              // 24*24*6
#define NTILE 216               // NCELL/16
#define SORTN 4096              // next pow2 >= NCELL

// per-batch workspace layout (float slots)
#define OFF_PCONF  0
#define OFF_PX     (NCELL*1)
#define OFF_PY     (NCELL*2)
#define OFF_PZ     (NCELL*3)
#define OFF_PN2    (NCELL*4)
#define OFF_PR     (NCELL*5)    // 6N floats
#define OFF_TCONF  (NCELL*11)
#define OFF_TX     (NCELL*12)
#define OFF_TY     (NCELL*13)
#define OFF_TZ     (NCELL*14)
#define OFF_TN2    (NCELL*15)   // masked: tval ? |t|^2 : 2e18
#define OFF_TR     (NCELL*16)   // 6N floats
#define OFF_KEEP   (NCELL*22)   // int
#define OFF_TVAL   (NCELL*23)   // int
#define OFF_MIND2  (NCELL*24)   // float (squared distances, masked -> >=1e18)
#define OFF_MINIDX (NCELL*25)   // int
#define OFF_BEST   (NCELL*26)   // uint (float bits of d^2)
#define OFF_PCOL   (NCELL*27)   // float4 per pred col: (x, y, z, keep ? |p|^2 : 2e18)
#define BSTRIDE    (NCELL*31)

#define N2BIG   2.0e18f
#define C2      4.0f            // CUTOFF^2
#define CAND2   1.0f            // (CUTOFF/2)^2
#define ZHI     8.00001f

typedef __attribute__((ext_vector_type(2))) float v2f;
typedef __attribute__((ext_vector_type(8))) float v8f;

// ---------------- K1: decode + init ----------------
__global__ __launch_bounds__(256) void k1_decode(const float* __restrict__ preds,
                                                 const float* __restrict__ targs,
                                                 float* __restrict__ ws) {
    int gid = blockIdx.x * blockDim.x + threadIdx.x;
    if (gid >= NBATCH * NCELL) return;
    int b = gid / NCELL;
    int i = gid - b * NCELL;
    int iz = i % GZd;
    int iy = (i / GZd) % GYd;
    int ix = i / (GZd * GYd);
    float* base = ws + (size_t)b * BSTRIDE;
    const float sx = 25.0f / 24.0f, sy = 25.0f / 24.0f, sz = 4.0f / 6.0f;

    const float* gp = preds + ((size_t)b * NCELL + i) * NCH;
    {
        float conf = gp[0];
        float x = ((float)ix + gp[1]) * sx;
        float y = ((float)iy + gp[2]) * sy;
        float z = ((float)iz + gp[3]) * sz;
        base[OFF_PCONF + i] = conf;
        base[OFF_PX + i] = x; base[OFF_PY + i] = y; base[OFF_PZ + i] = z;
        base[OFF_PN2 + i] = x * x + y * y + z * z;
        #pragma unroll
        for (int k = 0; k < 6; ++k) base[OFF_PR + (size_t)i * 6 + k] = gp[4 + k];
        ((int*)(base + OFF_KEEP))[i] = (conf > 0.0f) ? 1 : 0;   // INV_SIG_TH = log(1) = 0
    }
    const float* gt = targs + ((size_t)b * NCELL + i) * NCH;
    {
        float conf = gt[0];
        float x = ((float)ix + gt[1]) * sx;
        float y = ((float)iy + gt[2]) * sy;
        float z = ((float)iz + gt[3]) * sz;
        int tval = (conf > 0.5f) ? 1 : 0;
        base[OFF_TCONF + i] = conf;
        base[OFF_TX + i] = x; base[OFF_TY + i] = y; base[OFF_TZ + i] = z;
        base[OFF_TN2 + i] = tval ? (x * x + y * y + z * z) : N2BIG; // fold tg_valid mask
        #pragma unroll
        for (int k = 0; k < 6; ++k) base[OFF_TR + (size_t)i * 6 + k] = gt[4 + k];
        ((int*)(base + OFF_TVAL))[i] = tval;
    }
    ((unsigned*)(base + OFF_BEST))[i] = __float_as_uint(1.0e18f);
}

// ---------------- K2: greedy NMS (one block per batch) ----------------
__global__ __launch_bounds__(512) void k2_nms(float* __restrict__ ws) {
    __shared__ float sc[SORTN];
    __shared__ int   si[SORTN];
    __shared__ unsigned char sk[NCELL];
    __shared__ float sBx, sBy, sBz;
    __shared__ int sAct, sIdx;

    int b = blockIdx.x;
    int tid = threadIdx.x, bd = blockDim.x;
    float* base = ws + (size_t)b * BSTRIDE;
    const float* PC = base + OFF_PCONF;
    const float* PX = base + OFF_PX;
    const float* PY = base + OFF_PY;
    const float* PZ = base + OFF_PZ;
    const float* PN2 = base + OFF_PN2;
    int* KEEP = (int*)(base + OFF_KEEP);

    for (int i = tid; i < SORTN; i += bd) {
        sc[i] = (i < NCELL) ? PC[i] : -3.4e38f;
        si[i] = i;
    }
    for (int i = tid; i < NCELL; i += bd) sk[i] = (unsigned char)(KEEP[i] != 0);
    __syncthreads();

    // bitonic sort, descending by conf, ascending index on ties (stable-equivalent)
    for (int k = 2; k <= SORTN; k <<= 1) {
        for (int j = k >> 1; j > 0; j >>= 1) {
            for (int i = tid; i < SORTN; i += bd) {
                int p = i ^ j;
                if (p > i) {
                    float a = sc[i], c = sc[p];
                    int ia = si[i], ic = si[p];
                    bool before = (a > c) || (a == c && ia < ic);
                    bool up = ((i & k) == 0);
                    if (up ? !before : before) {
                        sc[i] = c; sc[p] = a; si[i] = ic; si[p] = ia;
                    }
                }
            }
            __syncthreads();
        }
    }

    // sequential greedy suppression in sorted order
    for (int i = 0; i < NCELL; ++i) {
        if (tid == 0) {
            int idx = si[i];
            sIdx = idx;
            int act = sk[idx];
            sAct = act;
            if (act) { sBx = PX[idx]; sBy = PY[idx]; sBz = PZ[idx]; }
        }
        __syncthreads();
        if (sAct) {
            float bx = sBx, by = sBy, bz = sBz;
            int idx = sIdx;
            for (int j2 = tid; j2 < NCELL; j2 += bd) {
                if (j2 != idx && sk[j2]) {
                    float dx = PX[j2] - bx, dy = PY[j2] - by, dz = PZ[j2] - bz;
                    if (dx * dx + dy * dy + dz * dz < C2) sk[j2] = 0;
                }
            }
        }
        __syncthreads();
    }
    // write back keep + packed pred columns with keep folded into |p|^2
    float4* PCOL = (float4*)(base + OFF_PCOL);
    for (int i = tid; i < NCELL; i += bd) {
        int k = sk[i];
        KEEP[i] = k;
        float4 q;
        q.x = PX[i]; q.y = PY[i]; q.z = PZ[i];
        q.w = k ? PN2[i] : N2BIG;
        PCOL[i] = q;
    }
}

// ---------------- K3: WMMA distance tiles + per-target argmin ----------------
// grid = NBATCH*27 blocks, 256 threads (8 waves); one wave = one 16-target tile.
// Pred columns staged once per block into LDS (54KB of the WGP's 320KB).
__global__ __launch_bounds__(256) void k3_match(float* __restrict__ ws) {
    __shared__ float4 sP[NCELL];

    int b    = blockIdx.x / 27;
    int tblk = blockIdx.x % 27;
    int wave = threadIdx.x >> 5;
    int lane = threadIdx.x & 31;
    int m16  = lane & 15;
    int hf   = lane >> 4;          // half of the wave (K=0,1 vs K=2,3 for A)
    int tile = tblk * 8 + wave;    // 0..215
    int tbase = tile * 16;

    float* base = ws + (size_t)b * BSTRIDE;
    const float* TX = base + OFF_TX;  const float* TY = base + OFF_TY;
    const float* TZ = base + OFF_TZ;  const float* TN2 = base + OFF_TN2;

    // cooperative stage of packed pred columns: global_load_b128 -> ds_store_b128
    {
        const float4* PCOL = (const float4*)(base + OFF_PCOL);
        for (int i = threadIdx.x; i < NCELL; i += 256) sP[i] = PCOL[i];
    }
    __syncthreads();

    // A operand: 16x4 F32. Lanes 0-15: M=m16, K=0,1 ; lanes 16-31: M=m16, K=2,3 (K=3 zero pad).
    int ra = tbase + m16;
    v2f A;
    A.x = hf ? TZ[ra] : TX[ra];
    A.y = hf ? 0.0f   : TY[ra];

    // each lane owns accumulator rows M = hf*8 + r (r=0..7); tg_valid already folded into TN2
    int rowb = tbase + hf * 8;
    float tn2v[8];
    #pragma unroll
    for (int r = 0; r < 8; ++r) tn2v[r] = TN2[rowb + r];

    float rmin[8]; int ridx[8];
    #pragma unroll
    for (int r = 0; r < 8; ++r) { rmin[r] = 3.0e38f; ridx[r] = 0; }

    for (int pt = 0; pt < NTILE; ++pt) {
        int col = pt * 16 + m16;
        float4 q = sP[col];                     // ds_load_b128
        // B operand: 4x16 F32 (column per lane, symmetric to A layout)
        v2f Bv;
        Bv.x = hf ? q.z : q.x;
        Bv.y = hf ? 0.0f : q.y;
        float pn2c = q.w;                       // keep mask folded (>=2e18 if suppressed)

        v8f acc = {};
        acc = __builtin_amdgcn_wmma_f32_16x16x4_f32(
            /*neg_a=*/false, A, /*neg_b=*/false, Bv,
            /*c_mod=*/(short)0, acc, /*reuse_a=*/false, /*reuse_b=*/false);

        #pragma unroll
        for (int r = 0; r < 8; ++r) {
            float d2 = tn2v[r] + pn2c - 2.0f * acc[r];
            d2 = d2 < 0.0f ? 0.0f : d2;         // masked rows/cols stay >= 1e18
            if (d2 < rmin[r]) { rmin[r] = d2; ridx[r] = col; }
        }
    }

    // cross-lane argmin over the 16 lanes of each half (N residue classes)
    #pragma unroll
    for (int off = 1; off < 16; off <<= 1) {
        #pragma unroll
        for (int r = 0; r < 8; ++r) {
            float om = __shfl_xor(rmin[r], off, 32);
            int   oi = __shfl_xor(ridx[r], off, 32);
            if (om < rmin[r] || (om == rmin[r] && oi < ridx[r])) { rmin[r] = om; ridx[r] = oi; }
        }
    }

    if (m16 == 0) {
        float* MIND2 = base + OFF_MIND2;
        int* MINIDX = (int*)(base + OFF_MINIDX);
        unsigned* BEST = (unsigned*)(base + OFF_BEST);
        #pragma unroll
        for (int r = 0; r < 8; ++r) {
            int row = rowb + r;
            MIND2[row] = rmin[r];
            MINIDX[row] = ridx[r];
            if (rmin[r] < CAND2)
                atomicMin(&BEST[ridx[r]], __float_as_uint(rmin[r]));
        }
    }
}

// ---------------- K4: matches, angles, bucketed reduction ----------------
__global__ __launch_bounds__(256) void k4_final(const float* __restrict__ ws,
                                                float* __restrict__ out) {
    __shared__ float s_tp[2], s_p[2], s_t[2], s_ang, s_nm;
    int b = blockIdx.x, tid = threadIdx.x;
    const float* base = ws + (size_t)b * BSTRIDE;
    const float* TZv = base + OFF_TZ;
    const float* PZv = base + OFF_PZ;
    const float* MIND2 = base + OFF_MIND2;
    const int* MINIDX = (const int*)(base + OFF_MINIDX);
    const unsigned* BEST = (const unsigned*)(base + OFF_BEST);
    const int* KEEP = (const int*)(base + OFF_KEEP);
    const int* TVAL = (const int*)(base + OFF_TVAL);
    const float* PR = base + OFF_PR;
    const float* TR = base + OFF_TR;

    if (tid == 0) {
        s_tp[0] = s_tp[1] = s_p[0] = s_p[1] = s_t[0] = s_t[1] = 0.0f;
        s_ang = 0.0f; s_nm = 0.0f;
    }
    __syncthreads();

    float tp[2] = {0, 0}, pc[2] = {0, 0}, tc[2] = {0, 0}, ang = 0.0f, nm = 0.0f;
    for (int i = tid; i < NCELL; i += blockDim.x) {
        float tz = TZv[i];
        int bz = -1;
        if (tz >= 0.0f && tz < 4.0f) bz = 0;
        else if (tz >= 4.0f && tz < ZHI) bz = 1;
        if (TVAL[i] && bz >= 0) tc[bz] += 1.0f;

        float pz = PZv[i];
        int bp = -1;
        if (pz >= 0.0f && pz < 4.0f) bp = 0;
        else if (pz >= 4.0f && pz < ZHI) bp = 1;
        if (KEEP[i] && bp >= 0) pc[bp] += 1.0f;

        float d2 = MIND2[i];
        int j = MINIDX[i];
        bool cand = d2 < CAND2;
        float bj = __uint_as_float(BEST[j]);
        bool m = cand && (d2 <= bj);
        if (m) {
            nm += 1.0f;
            if (bz >= 0) tp[bz] += 1.0f;
            const float* Rp = PR + (size_t)j * 6;
            const float* Rt = TR + (size_t)i * 6;
            #pragma unroll
            for (int r = 0; r < 2; ++r) {
                float dt = 0.0f, n1 = 0.0f, n2 = 0.0f;
                #pragma unroll
                for (int k = 0; k < 3; ++k) {
                    float a = Rp[r * 3 + k], c = Rt[r * 3 + k];
                    dt += a * c; n1 += a * a; n2 += c * c;
                }
                float cs = dt / (sqrtf(n1) * sqrtf(n2));
                float av;
                if (cs != cs) av = 90.0f;                       // nan -> 90
                else av = acosf(fminf(fmaxf(cs, -1.0f), 1.0f)) * 57.29577951308232f;
                ang += av;
            }
        }
    }
    atomicAdd(&s_tp[0], tp[0]); atomicAdd(&s_tp[1], tp[1]);
    atomicAdd(&s_p[0], pc[0]);  atomicAdd(&s_p[1], pc[1]);
    atomicAdd(&s_t[0], tc[0]);  atomicAdd(&s_t[1], tc[1]);
    atomicAdd(&s_ang, ang);     atomicAdd(&s_nm, nm);
    __syncthreads();

    if (tid == 0) {
        float nmv = s_nm;
        float am = (nmv > 0.0f) ? s_ang / (2.0f * fmaxf(nmv, 1.0f)) : 90.0f;
        #pragma unroll
        for (int row = 0; row < 2; ++row) {
            out[(size_t)b * 8 + row * 4 + 0] = s_tp[row];
            out[(size_t)b * 8 + row * 4 + 1] = s_p[row] - s_tp[row];
            out[(size_t)b * 8 + row * 4 + 2] = s_t[row] - s_tp[row];
            out[(size_t)b * 8 + row * 4 + 3] = am;
        }
    }
}

// ---------------- launcher ----------------
extern "C" void kernel_launch(void* const* d_in, const int* in_sizes, int n_in,
                              void* d_out, int out_size, void* d_ws, size_t ws_size,
                              hipStream_t stream) {
    const float* preds = (const float*)d_in[0];
    const float* targs = (const float*)d_in[1];
    float* ws = (float*)d_ws;
    float* out = (float*)d_out;

    int total = NBATCH * NCELL;
    k1_decode<<<(total + 255) / 256, 256, 0, stream>>>(preds, targs, ws);
    k2_nms<<<NBATCH, 512, 0, stream>>>(ws);
    k3_match<<<NBATCH * 27, 256, 0, stream>>>(ws);
    k4_final<<<NBATCH, 256, 0, stream>>>(ws, out);
}